// RGCN_16209206575328
// MI455X (gfx1250) — compile-verified
//
#include <hip/hip_runtime.h>

typedef __attribute__((ext_vector_type(2))) float v2f;
typedef __attribute__((ext_vector_type(8))) float v8f;

#define RGCN_R 7
#define RGCN_I 128
#define RGCN_O 64
#define RGCN_T 4

// ---------------- zero int buffer ----------------
__global__ void rgcn_zero_i32(int* __restrict__ p, int n) {
  int i = blockIdx.x * blockDim.x + threadIdx.x;
  if (i < n) p[i] = 0;
}

// ---------------- per-(dst, rel) edge counts ----------------
__global__ void rgcn_count(const int* __restrict__ dst, const int* __restrict__ etype,
                           int* __restrict__ cnt, int E) {
  int e = blockIdx.x * blockDim.x + threadIdx.x;
  if (e < E) atomicAdd(&cnt[dst[e] * RGCN_R + etype[e]], 1);
}

// ---------------- h[r] = x @ W_rel[r]  via V_WMMA_F32_16X16X4_F32 ----------------
// One wave computes a 16-node x 64-col tile. blockDim=256 (8 waves), grid.y = relation.
__global__ void __launch_bounds__(256) rgcn_gemm_h(const float* __restrict__ x,
                                                   const float* __restrict__ W_rel,
                                                   float* __restrict__ h, int n_nodes) {
  __shared__ float lds[8 * 16 * 68];        // per-wave 16x64 tile, stride 68 (bank-conflict pad)
  const int lane  = threadIdx.x & 31;
  const int wv    = threadIdx.x >> 5;
  const int node0 = (blockIdx.x * 8 + wv) * 16;
  const int r     = blockIdx.y;
  const int col   = lane & 15;
  const int khalf = (lane >> 4) << 1;       // 0 or 2  (A/B K-split across lane halves)
  const int mbase = (lane >> 4) << 3;       // 0 or 8  (C/D row split across lane halves)

  if (node0 < n_nodes) {
    const float* __restrict__ xr = x + (size_t)(node0 + col) * RGCN_I;
    const float* __restrict__ W  = W_rel + (size_t)r * RGCN_I * RGCN_O;
    v8f acc0 = {}, acc1 = {}, acc2 = {}, acc3 = {};
#pragma unroll 4
    for (int k = 0; k < RGCN_I; k += 4) {
      v2f a;
      a.x = xr[k + khalf];
      a.y = xr[k + khalf + 1];
      const float* Wk = W + (size_t)(k + khalf) * RGCN_O + col;
      v2f b0, b1, b2, b3;
      b0.x = Wk[0];  b0.y = Wk[RGCN_O + 0];
      b1.x = Wk[16]; b1.y = Wk[RGCN_O + 16];
      b2.x = Wk[32]; b2.y = Wk[RGCN_O + 32];
      b3.x = Wk[48]; b3.y = Wk[RGCN_O + 48];
      acc0 = __builtin_amdgcn_wmma_f32_16x16x4_f32(false, a, false, b0, (short)0, acc0, false, false);
      acc1 = __builtin_amdgcn_wmma_f32_16x16x4_f32(false, a, false, b1, (short)0, acc1, false, false);
      acc2 = __builtin_amdgcn_wmma_f32_16x16x4_f32(false, a, false, b2, (short)0, acc2, false, false);
      acc3 = __builtin_amdgcn_wmma_f32_16x16x4_f32(false, a, false, b3, (short)0, acc3, false, false);
    }
    float* t = &lds[wv * 16 * 68];
#pragma unroll
    for (int j = 0; j < 8; ++j) {
      t[(mbase + j) * 68 +  0 + col] = acc0[j];
      t[(mbase + j) * 68 + 16 + col] = acc1[j];
      t[(mbase + j) * 68 + 32 + col] = acc2[j];
      t[(mbase + j) * 68 + 48 + col] = acc3[j];
    }
  }
  __syncthreads();
  if (node0 < n_nodes) {
    const float* t = &lds[wv * 16 * 68];
    float* __restrict__ dp = h + ((size_t)r * n_nodes + node0) * RGCN_O;
#pragma unroll
    for (int i = 0; i < 8; ++i) {
      int idx = (lane + i * 32) << 2;       // float4 index into 16x64 tile
      int row = idx >> 6, cc = idx & 63;
      *(float4*)(dp + idx) = *(const float4*)(t + row * 68 + cc);   // 272B rows keep 16B align
    }
  }
}

// ---------------- out = (x @ W_root[node_type]) + b_root[node_type] ----------------
// Compute all 4 type-GEMMs per tile, select per node by its type. Fully writes out.
__global__ void __launch_bounds__(256) rgcn_root(const float* __restrict__ x,
                                                 const float* __restrict__ W_root,
                                                 const float* __restrict__ b_root,
                                                 const int* __restrict__ node_type,
                                                 float* __restrict__ out, int n_nodes) {
  __shared__ float lds[8 * 16 * 68];
  const int lane  = threadIdx.x & 31;
  const int wv    = threadIdx.x >> 5;
  const int node0 = (blockIdx.x * 8 + wv) * 16;
  const int col   = lane & 15;
  const int khalf = (lane >> 4) << 1;
  const int mbase = (lane >> 4) << 3;

  if (node0 < n_nodes) {
    int nt[8];
#pragma unroll
    for (int j = 0; j < 8; ++j) nt[j] = node_type[node0 + mbase + j];

    const float* __restrict__ xr = x + (size_t)(node0 + col) * RGCN_I;
    v8f fin0 = {}, fin1 = {}, fin2 = {}, fin3 = {};
    for (int t = 0; t < RGCN_T; ++t) {
      const float* __restrict__ W = W_root + (size_t)t * RGCN_I * RGCN_O;
      v8f a0 = {}, a1 = {}, a2 = {}, a3 = {};
#pragma unroll 4
      for (int k = 0; k < RGCN_I; k += 4) {
        v2f a;
        a.x = xr[k + khalf];
        a.y = xr[k + khalf + 1];
        const float* Wk = W + (size_t)(k + khalf) * RGCN_O + col;
        v2f b0, b1, b2, b3;
        b0.x = Wk[0];  b0.y = Wk[RGCN_O + 0];
        b1.x = Wk[16]; b1.y = Wk[RGCN_O + 16];
        b2.x = Wk[32]; b2.y = Wk[RGCN_O + 32];
        b3.x = Wk[48]; b3.y = Wk[RGCN_O + 48];
        a0 = __builtin_amdgcn_wmma_f32_16x16x4_f32(false, a, false, b0, (short)0, a0, false, false);
        a1 = __builtin_amdgcn_wmma_f32_16x16x4_f32(false, a, false, b1, (short)0, a1, false, false);
        a2 = __builtin_amdgcn_wmma_f32_16x16x4_f32(false, a, false, b2, (short)0, a2, false, false);
        a3 = __builtin_amdgcn_wmma_f32_16x16x4_f32(false, a, false, b3, (short)0, a3, false, false);
      }
      float bb0 = b_root[t * RGCN_O +  0 + col];
      float bb1 = b_root[t * RGCN_O + 16 + col];
      float bb2 = b_root[t * RGCN_O + 32 + col];
      float bb3 = b_root[t * RGCN_O + 48 + col];
#pragma unroll
      for (int j = 0; j < 8; ++j) {
        if (nt[j] == t) {
          fin0[j] = a0[j] + bb0;
          fin1[j] = a1[j] + bb1;
          fin2[j] = a2[j] + bb2;
          fin3[j] = a3[j] + bb3;
        }
      }
    }
    float* t = &lds[wv * 16 * 68];
#pragma unroll
    for (int j = 0; j < 8; ++j) {
      t[(mbase + j) * 68 +  0 + col] = fin0[j];
      t[(mbase + j) * 68 + 16 + col] = fin1[j];
      t[(mbase + j) * 68 + 32 + col] = fin2[j];
      t[(mbase + j) * 68 + 48 + col] = fin3[j];
    }
  }
  __syncthreads();
  if (node0 < n_nodes) {
    const float* t = &lds[wv * 16 * 68];
    float* __restrict__ dp = out + (size_t)node0 * RGCN_O;
#pragma unroll
    for (int i = 0; i < 8; ++i) {
      int idx = (lane + i * 32) << 2;
      int row = idx >> 6, cc = idx & 63;
      *(float4*)(dp + idx) = *(const float4*)(t + row * 68 + cc);
    }
  }
}

// ---------------- edge scatter: out[dst] += h[rel, src] / cnt[dst,rel] ----------------
// 16 threads per edge, float4 message chunks, fp32 atomics (out is L2-resident: 77MB).
__global__ void rgcn_scatter(const int* __restrict__ src, const int* __restrict__ dst,
                             const int* __restrict__ etype, const int* __restrict__ cnt,
                             const float* __restrict__ h, float* __restrict__ out,
                             int E, int n_nodes) {
  long long gid = (long long)blockIdx.x * blockDim.x + threadIdx.x;
  int e = (int)(gid >> 4);
  if (e >= E) return;
  int c4 = ((int)gid & 15) << 2;
  int d = dst[e], s = src[e], r = etype[e];
  float w = 1.0f / fmaxf((float)cnt[d * RGCN_R + r], 1.0f);
  const float4 v = *(const float4*)(h + ((size_t)r * n_nodes + s) * RGCN_O + c4);
  float* o = out + (size_t)d * RGCN_O + c4;
  atomicAdd(o + 0, v.x * w);
  atomicAdd(o + 1, v.y * w);
  atomicAdd(o + 2, v.z * w);
  atomicAdd(o + 3, v.w * w);
}

extern "C" void kernel_launch(void* const* d_in, const int* in_sizes, int n_in,
                              void* d_out, int out_size, void* d_ws, size_t ws_size,
                              hipStream_t stream) {
  const float* x      = (const float*)d_in[0];
  const int*   eidx   = (const int*)d_in[1];
  const int*   etype  = (const int*)d_in[2];
  const int*   ntype  = (const int*)d_in[3];
  const float* W_rel  = (const float*)d_in[4];
  const float* W_root = (const float*)d_in[5];
  const float* b_root = (const float*)d_in[6];
  float* out = (float*)d_out;

  const int N = in_sizes[0] / RGCN_I;   // 300000
  const int E = in_sizes[2];            // 4000000
  const int* src = eidx;                // edge_index[0]
  const int* dst = eidx + E;            // edge_index[1]

  // workspace: h[R,N,O] (537.6 MB) then cnt[N*R] (8.4 MB)
  float* h  = (float*)d_ws;
  int*   cnt = (int*)((char*)d_ws + (size_t)RGCN_R * N * RGCN_O * sizeof(float));

  rgcn_zero_i32<<<(N * RGCN_R + 255) / 256, 256, 0, stream>>>(cnt, N * RGCN_R);
  rgcn_count<<<(E + 255) / 256, 256, 0, stream>>>(dst, etype, cnt, E);

  int tiles  = (N + 15) / 16;
  int blocks = (tiles + 7) / 8;
  rgcn_gemm_h<<<dim3(blocks, RGCN_R), 256, 0, stream>>>(x, W_rel, h, N);
  rgcn_root<<<blocks, 256, 0, stream>>>(x, W_root, b_root, ntype, out, N);

  long long sthreads = (long long)E * 16;
  rgcn_scatter<<<(unsigned)((sthreads + 255) / 256), 256, 0, stream>>>(
      src, dst, etype, cnt, h, out, E, N);
}